// RetrievalLoss_33217277067289
// MI455X (gfx1250) — compile-verified
//
#include <hip/hip_runtime.h>

#define N_PTS 2048
#define DIM   128
#define DELTA_F 1.0f

typedef float v2f __attribute__((ext_vector_type(2)));
typedef float v8f __attribute__((ext_vector_type(8)));

// ---------------------------------------------------------------------------
// Kernel 1: per-row squared norms  sq[i] = ||Q[i,:]||^2
// ---------------------------------------------------------------------------
__global__ __launch_bounds__(256)
void norms_kernel(const float* __restrict__ q, float* __restrict__ sq) {
    int i = blockIdx.x * blockDim.x + threadIdx.x;
    if (i >= N_PTS) return;
    const float4* row = (const float4*)(q + (size_t)i * DIM);
    float s = 0.f;
#pragma unroll
    for (int k = 0; k < DIM / 4; ++k) {
        float4 v = row[k];
        s += v.x * v.x + v.y * v.y + v.z * v.z + v.w * v.w;
    }
    sq[i] = s;
}

// ---------------------------------------------------------------------------
// Kernel 2: fused QQ^T (WMMA f32 16x16x4) + hard positive/negative mining.
// Block = 16-row strip, 4 waves; wave w sweeps column tiles [32w, 32w+32).
// A fragments hoisted to registers; B stream double-buffered in 2 K-chunks
// so loads for the next chunk/tile are always in flight under the WMMA chain.
// ---------------------------------------------------------------------------
__global__ __launch_bounds__(128)
void hardmine_kernel(const float* __restrict__ q, const int* __restrict__ tgt,
                     const float* __restrict__ sq, float* __restrict__ lossRow) {
    const int lane = threadIdx.x & 31;
    const int wave = threadIdx.x >> 5;   // 0..3
    const int half = lane >> 4;          // 0 or 1 (lane group)
    const int c16  = lane & 15;
    const int rowbase = blockIdx.x * 16;

    // ---- Row-side constants for the 8 accumulator slots this lane covers:
    // acc[v] holds element (m = rowbase + v + 8*half, n = colbase + c16)
    float sqRow[8]; int tRow[8];
#pragma unroll
    for (int v = 0; v < 8; ++v) {
        int m = rowbase + v + 8 * half;
        sqRow[v] = sq[m];
        tRow[v]  = tgt[m];
    }

    // ---- Running argmax (pos) / argmin (neg) per row-slot.
    float pS[8], pD[8], nS[8];
    int   pN[8], nN[8];
#pragma unroll
    for (int v = 0; v < 8; ++v) {
        pS[v] = -__builtin_inff(); pD[v] = 0.f; pN[v] = 0x7fffffff;
        nS[v] =  __builtin_inff();               nN[v] = 0x7fffffff;
    }

    // ---- Hoist A fragments (loop-invariant across all column tiles).
    // Lane L holds A[m = L%16, k = 4s + 2*(L/16) + {0,1}].
    const float* arow = q + (size_t)(rowbase + c16) * DIM + 2 * half;
    v2f afrag[32];
#pragma unroll
    for (int s = 0; s < 32; ++s) afrag[s] = *(const v2f*)(arow + 4 * s);

    const int jtiles = (N_PTS / 16) / 4;      // 32 tiles per wave
    const int j0 = wave * jtiles;

    // B fragment pointer for a given column tile (same lane layout as A).
    auto bptr = [&](int jt) {
        return q + (size_t)((j0 + jt) * 16 + c16) * DIM + 2 * half;
    };

    v2f b0[16], b1[16];   // K-chunk double buffer (chunk = 16 WMMA steps)

    // Prologue: chunk 0 of tile 0.
    {
        const float* br = bptr(0);
#pragma unroll
        for (int s = 0; s < 16; ++s) b0[s] = *(const v2f*)(br + 4 * s);
    }

#pragma unroll 1
    for (int jt = 0; jt < jtiles; ++jt) {
        const int colbase = (j0 + jt) * 16;
        const int n       = colbase + c16;

        // Epilogue operands: issue early so they are resident by mask time.
        const float sqCol = sq[n];
        const int   tCol  = tgt[n];

        const float* brCur = bptr(jt);
        const int jnext    = (jt + 1 < jtiles) ? jt + 1 : 0;  // branchless guard
        const float* brNxt = bptr(jnext);

        v8f acc = {0.f, 0.f, 0.f, 0.f, 0.f, 0.f, 0.f, 0.f};

        // Prefetch chunk 1 of this tile, then run chunk-0 WMMA chain.
#pragma unroll
        for (int s = 0; s < 16; ++s) b1[s] = *(const v2f*)(brCur + 64 + 4 * s);
#pragma unroll
        for (int s = 0; s < 16; ++s)
            acc = __builtin_amdgcn_wmma_f32_16x16x4_f32(
                false, afrag[s], false, b0[s], (short)0, acc, false, false);

        // Prefetch chunk 0 of the next tile, then run chunk-1 WMMA chain.
#pragma unroll
        for (int s = 0; s < 16; ++s) b0[s] = *(const v2f*)(brNxt + 4 * s);
#pragma unroll
        for (int s = 0; s < 16; ++s)
            acc = __builtin_amdgcn_wmma_f32_16x16x4_f32(
                false, afrag[16 + s], false, b1[s], (short)0, acc, false, false);

        // ---- Epilogue: dist = sq[m] + sq[n] - 2*dot, mask, update best.
#pragma unroll
        for (int v = 0; v < 8; ++v) {
            int   m    = rowbase + v + 8 * half;
            float dist = sqRow[v] + sqCol - 2.0f * acc[v];
            bool  same = (tRow[v] == tCol);
            bool  diag = (m == n);
            float ps = diag ? -__builtin_inff() : (same ? dist : 0.0f);
            float ns = (same || diag) ? __builtin_inff() : dist;
            if (ps > pS[v]) { pS[v] = ps; pN[v] = n; pD[v] = dist; }
            if (ns < nS[v]) { nS[v] = ns; nN[v] = n; }
        }
    }

    // ---- Butterfly reduce across the 16 lanes of each half (tile columns).
    // Tie-break: smaller index wins (matches jnp first-occurrence semantics).
#pragma unroll
    for (int v = 0; v < 8; ++v) {
#pragma unroll
        for (int msk = 1; msk < 16; msk <<= 1) {
            float oS = __shfl_xor(pS[v], msk, 32);
            int   oN = __shfl_xor(pN[v], msk, 32);
            float oD = __shfl_xor(pD[v], msk, 32);
            bool take = (oS > pS[v]) || (oS == pS[v] && oN < pN[v]);
            pS[v] = take ? oS : pS[v];
            pN[v] = take ? oN : pN[v];
            pD[v] = take ? oD : pD[v];

            float uS = __shfl_xor(nS[v], msk, 32);
            int   uN = __shfl_xor(nN[v], msk, 32);
            bool tk2 = (uS < nS[v]) || (uS == nS[v] && uN < nN[v]);
            nS[v] = tk2 ? uS : nS[v];
            nN[v] = tk2 ? uN : nN[v];
        }
    }

    // ---- Combine the 4 waves' candidates per row via LDS (ascending-j order).
    __shared__ float sPS[4][16], sPD[4][16], sNS[4][16];
    __shared__ int   sPN[4][16], sNN[4][16];
    if (c16 == 0) {
#pragma unroll
        for (int v = 0; v < 8; ++v) {
            int r = v + 8 * half;
            sPS[wave][r] = pS[v]; sPN[wave][r] = pN[v]; sPD[wave][r] = pD[v];
            sNS[wave][r] = nS[v]; sNN[wave][r] = nN[v];
        }
    }
    __syncthreads();

    if (threadIdx.x < 16) {
        int r = threadIdx.x;
        float bPS = -__builtin_inff(), bPD = 0.f, bNS = __builtin_inff();
        int   bPN = 0x7fffffff, bNN = 0x7fffffff;
#pragma unroll
        for (int w = 0; w < 4; ++w) {
            float s0 = sPS[w][r]; int n0 = sPN[w][r];
            if (s0 > bPS || (s0 == bPS && n0 < bPN)) { bPS = s0; bPN = n0; bPD = sPD[w][r]; }
            float s1 = sNS[w][r]; int n1 = sNN[w][r];
            if (s1 < bNS || (s1 == bNS && n1 < bNN)) { bNS = s1; bNN = n1; }
        }
        // bPD = d(q, hard_pos); bNS = d(q, hard_neg)
        float loss = DELTA_F - bPD + bNS;
        lossRow[rowbase + r] = loss > 0.f ? loss : 0.f;
    }
}

// ---------------------------------------------------------------------------
// Kernel 3: deterministic fixed-order mean of 2048 per-row losses.
// ---------------------------------------------------------------------------
__global__ __launch_bounds__(256)
void reduce_kernel(const float* __restrict__ lossRow, float* __restrict__ out) {
    __shared__ float buf[256];
    float s = 0.f;
    for (int i = threadIdx.x; i < N_PTS; i += 256) s += lossRow[i];
    buf[threadIdx.x] = s;
    __syncthreads();
    for (int step = 128; step > 0; step >>= 1) {
        if ((int)threadIdx.x < step) buf[threadIdx.x] += buf[threadIdx.x + step];
        __syncthreads();
    }
    if (threadIdx.x == 0) out[0] = buf[0] * (1.0f / N_PTS);
}

// ---------------------------------------------------------------------------
extern "C" void kernel_launch(void* const* d_in, const int* in_sizes, int n_in,
                              void* d_out, int out_size, void* d_ws, size_t ws_size,
                              hipStream_t stream) {
    const float* q   = (const float*)d_in[0];   // queries [2048,128] f32
    const int*   tgt = (const int*)d_in[1];     // targets [2048] int
    float* ws      = (float*)d_ws;
    float* sq      = ws;                        // 2048 floats
    float* lossRow = ws + N_PTS;                // 2048 floats

    norms_kernel   <<<N_PTS / 256, 256, 0, stream>>>(q, sq);
    hardmine_kernel<<<N_PTS / 16, 128, 0, stream>>>(q, tgt, sq, lossRow);
    reduce_kernel  <<<1, 256, 0, stream>>>(lossRow, (float*)d_out);
}